// SegmentationLoss_10737418240401
// MI455X (gfx1250) — compile-verified
//
#include <hip/hip_runtime.h>

typedef __attribute__((ext_vector_type(2))) float v2f;
typedef __attribute__((ext_vector_type(4))) float v4f;
typedef __attribute__((ext_vector_type(8))) float v8f;

// Problem geometry (fixed by the reference setup)
#define BS    8
#define NF    16
#define NI    8
#define HW    65536                 // 256*256, per-plane elements
#define NLOC  (BS * HW)             // 524288 locations
#define NQUAD (NLOC / 4)            // 131072 (each thread handles 4 pixels)
#define TPB   256
#define NBLK  (NQUAD / TPB)         // 512 blocks, exact cover

__global__ __launch_bounds__(TPB)
void seg_loss_partial(const float* __restrict__ input,
                      const float* __restrict__ target,
                      float* __restrict__ partial)
{
    const int tid = blockIdx.x * TPB + threadIdx.x;   // 0 .. NQUAD-1 exactly
    const int loc = tid << 2;                         // 4 consecutive pixels
    const int b   = loc >> 16;                        // image index
    const int li  = loc & 0xFFFF;                     // pixel within plane

    // ---- stream pred planes: 16 coalesced b128 NT loads -> S1, S2 ----
    const v4f* __restrict__ pb =
        (const v4f*)(input + (size_t)b * NF * HW + li);
    v4f S1 = (v4f)0.0f, S2 = (v4f)0.0f;
#pragma unroll
    for (int f = 0; f < NF; ++f) {
        v4f p = __builtin_nontemporal_load(&pb[(size_t)f * (HW / 4)]);
        S1 += p;
        S2 += p * p;
    }

    // ---- stream gt planes: 8 coalesced b128 NT loads, accumulate g*d ----
    const v4f* __restrict__ gb =
        (const v4f*)(target + (size_t)b * NI * HW + li);
    float acc01 = 0.0f, acc23 = 0.0f;   // two accumulators -> fills 16x4 A
#pragma unroll
    for (int i = 0; i < NI; ++i) {
        v4f g = __builtin_nontemporal_load(&gb[(size_t)i * (HW / 4)]);
        v4f t = S2 - 2.0f * g * S1 + (float)NF * g * g;  // = sum_f (p_f - g)^2
        float d0 = __builtin_amdgcn_sqrtf(fmaxf(t.x, 0.0f));
        float d1 = __builtin_amdgcn_sqrtf(fmaxf(t.y, 0.0f));
        float d2 = __builtin_amdgcn_sqrtf(fmaxf(t.z, 0.0f));
        float d3 = __builtin_amdgcn_sqrtf(fmaxf(t.w, 0.0f));
        acc01 += g.x * d0 + g.y * d1;
        acc23 += g.z * d2 + g.w * d3;
    }

    // ---- wave reduction via V_WMMA_F32_16X16X4_F32 (full f32 precision) ----
    // A (16x4): 2 VGPRs/lane hold the 64 per-wave accumulators.
    // B (4x16) = ones  =>  D[m][n] = rowsum_m(A).
    v2f a = {acc01, acc23};
    v2f one = {1.0f, 1.0f};
    v8f c = {};
    v8f d = __builtin_amdgcn_wmma_f32_16x16x4_f32(
        /*neg_a=*/false, a, /*neg_b=*/false, one,
        /*c_mod=*/(short)0, c, /*reuse_a=*/false, /*reuse_b=*/false);

    // Lane L<16 holds D[0..7][L] in its 8 regs; lane 16+L holds D[8..15][L].
    float v = d[0] + d[1] + d[2] + d[3] + d[4] + d[5] + d[6] + d[7];
    float v16 = __shfl(v, 16, 32);        // lane 16's partial (M=8..15 rowsums)
    float wave_total = v + v16;           // valid on lane 0: sum of all 64 accs

    __shared__ float lds[TPB / 32];
    const int lane = threadIdx.x & 31;
    const int wave = threadIdx.x >> 5;
    if (lane == 0) lds[wave] = wave_total;
    __syncthreads();
    if (threadIdx.x == 0) {
        float s = 0.0f;
#pragma unroll
        for (int w = 0; w < TPB / 32; ++w) s += lds[w];
        partial[blockIdx.x] = s;
    }
}

__global__ __launch_bounds__(TPB)
void seg_loss_final(const float* __restrict__ partial,
                    float* __restrict__ out, int n)
{
    __shared__ float l[TPB];
    float s = 0.0f;
    for (int i = threadIdx.x; i < n; i += TPB) s += partial[i];
    l[threadIdx.x] = s;
    __syncthreads();
#pragma unroll
    for (int off = TPB / 2; off > 0; off >>= 1) {
        if (threadIdx.x < off) l[threadIdx.x] += l[threadIdx.x + off];
        __syncthreads();
    }
    if (threadIdx.x == 0) out[0] = l[0];
}

extern "C" void kernel_launch(void* const* d_in, const int* in_sizes, int n_in,
                              void* d_out, int out_size, void* d_ws, size_t ws_size,
                              hipStream_t stream) {
    const float* input  = (const float*)d_in[0];   // (8,16,256,256) f32
    const float* target = (const float*)d_in[1];   // (8, 8,256,256) f32
    // d_in[2] (n_objects) and d_in[3] (max_n_objects) are unused by the math.
    float* partial = (float*)d_ws;                 // NBLK floats of scratch
    float* out     = (float*)d_out;                // single f32 scalar

    seg_loss_partial<<<NBLK, TPB, 0, stream>>>(input, target, partial);
    seg_loss_final<<<1, TPB, 0, stream>>>(partial, out, NBLK);
}